// MNL_3478923510573
// MI455X (gfx1250) — compile-verified
//
#include <hip/hip_runtime.h>
#include <hip/hip_bf16.h>
#include <cstdint>
#include <cstddef>

// Problem geometry (fixed by the reference): rows of N=4096 options.
#define OPT_N   4096
#define BLOCK   256
#define V4_ROW  (OPT_N / 4)        // 1024 float4 per row
#define CHUNKS  (V4_ROW / BLOCK)   // 4 float4 per thread per row
#define NWAVES  (BLOCK / 32)       // 8 wave32s per block
#define GRID_MAX 4096              // persistent blocks; 4 rows each at B=16384

// Native clang vector (HIP's float4 is a class and is rejected by the
// nontemporal builtins).
typedef float v4f __attribute__((ext_vector_type(4)));

// Low 32 bits of a generic pointer into LDS == byte offset from the wave's
// LDS base (the LDS aperture uses the address's low 32 bits).
__device__ __forceinline__ unsigned lds_addr32(const void* p) {
  return (unsigned)(unsigned long long)p;
}

// CDNA5 async DMA: global -> LDS, 16 bytes per lane, tracked by ASYNCcnt.
__device__ __forceinline__ void async_g2l_b128(unsigned lds_off, const void* gaddr) {
  asm volatile("global_load_async_to_lds_b128 %0, %1, off"
               :
               : "v"(lds_off), "v"(gaddr)
               : "memory");
}

#define S_WAIT_ASYNCCNT(N) asm volatile("s_wait_asynccnt " #N ::: "memory")

__global__ void __launch_bounds__(BLOCK)
mnl_masked_softmax_kernel(const float* __restrict__ e,
                          const float* __restrict__ diag,
                          float* __restrict__ out_softmax,
                          float* __restrict__ out_scaled,
                          int B, int nblocks)
{
  __shared__ __align__(16) float ebuf[2][OPT_N];   // 32 KB double buffer
  __shared__ float red_max[NWAVES];
  __shared__ float red_sum[NWAVES];

  const int tid  = threadIdx.x;
  const int lane = tid & 31;
  const int wid  = tid >> 5;
  const unsigned lds_base = lds_addr32(&ebuf[0][0]);
  const float NEG_INF = -__builtin_inff();

  // diag is reused by every row this thread touches (same column set):
  // keep it resident in 16 VGPRs for the whole block lifetime.
  const v4f* dg4 = (const v4f*)diag;
  v4f dg[CHUNKS];
#pragma unroll
  for (int t = 0; t < CHUNKS; ++t) dg[t] = dg4[tid + t * BLOCK];

  int row = blockIdx.x;
  if (row >= B) return;

  // Prefetch first row into buffer 0 (4 async b128 per thread).
  {
    const float* src = e + (size_t)row * OPT_N;
#pragma unroll
    for (int t = 0; t < CHUNKS; ++t) {
      int v4 = tid + t * BLOCK;
      async_g2l_b128(lds_base + (unsigned)(v4 * 16),
                     (const void*)(src + v4 * 4));
    }
  }

  int buf = 0;
  for (; row < B; row += nblocks, buf ^= 1) {
    // ---- issue async DMA for the next row into the other buffer ----
    const int nrow = row + nblocks;
    if (nrow < B) {
      const float* src = e + (size_t)nrow * OPT_N;
      const unsigned nb_off = lds_base + (unsigned)((buf ^ 1) * OPT_N * 4);
#pragma unroll
      for (int t = 0; t < CHUNKS; ++t) {
        int v4 = tid + t * BLOCK;
        async_g2l_b128(nb_off + (unsigned)(v4 * 16),
                       (const void*)(src + v4 * 4));
      }
      // 4 in flight (next row); the current row's 4, issued earlier, are done.
      S_WAIT_ASYNCCNT(4);
    } else {
      S_WAIT_ASYNCCNT(0);
    }
    __syncthreads();  // everyone's slice of ebuf[buf] is resident

    // ---- pass 1: read row from LDS, masked values + e_scaled + local max ----
    const v4f* eb4 = (const v4f*)&ebuf[buf][0];
    v4f vv[CHUNKS];
    float lmax = NEG_INF;
    float* osc = out_scaled + (size_t)row * OPT_N;
#pragma unroll
    for (int t = 0; t < CHUNKS; ++t) {
      const int v4 = tid + t * BLOCK;
      v4f ev = eb4[v4];
      v4f es;  // e * diag, true multiply (preserves -0.0 like the reference)
      es.x = ev.x * dg[t].x;  es.y = ev.y * dg[t].y;
      es.z = ev.z * dg[t].z;  es.w = ev.w * dg[t].w;
      __builtin_nontemporal_store(es, (v4f*)(osc + v4 * 4));
      vv[t].x = (ev.x == 1.0f) ? dg[t].x : NEG_INF;
      vv[t].y = (ev.y == 1.0f) ? dg[t].y : NEG_INF;
      vv[t].z = (ev.z == 1.0f) ? dg[t].z : NEG_INF;
      vv[t].w = (ev.w == 1.0f) ? dg[t].w : NEG_INF;
      lmax = fmaxf(lmax, fmaxf(fmaxf(vv[t].x, vv[t].y), fmaxf(vv[t].z, vv[t].w)));
    }

    // ---- block reduce max (wave32 shfl + 8-entry LDS combine) ----
#pragma unroll
    for (int o = 16; o > 0; o >>= 1) lmax = fmaxf(lmax, __shfl_xor(lmax, o, 32));
    if (lane == 0) red_max[wid] = lmax;
    __syncthreads();
    float m = red_max[lane & (NWAVES - 1)];
#pragma unroll
    for (int o = NWAVES / 2; o > 0; o >>= 1) m = fmaxf(m, __shfl_xor(m, o, 32));

    // ---- pass 2: exp + local sum (values already in registers) ----
    float lsum = 0.0f;
#pragma unroll
    for (int t = 0; t < CHUNKS; ++t) {
      vv[t].x = __expf(vv[t].x - m);
      vv[t].y = __expf(vv[t].y - m);
      vv[t].z = __expf(vv[t].z - m);
      vv[t].w = __expf(vv[t].w - m);
      lsum += (vv[t].x + vv[t].y) + (vv[t].z + vv[t].w);
    }
#pragma unroll
    for (int o = 16; o > 0; o >>= 1) lsum += __shfl_xor(lsum, o, 32);
    if (lane == 0) red_sum[wid] = lsum;
    __syncthreads();
    float s = red_sum[lane & (NWAVES - 1)];
#pragma unroll
    for (int o = NWAVES / 2; o > 0; o >>= 1) s += __shfl_xor(s, o, 32);

    // ---- normalize + store softmax ----
    const float inv = 1.0f / s;
    float* osm = out_softmax + (size_t)row * OPT_N;
#pragma unroll
    for (int t = 0; t < CHUNKS; ++t) {
      const int v4 = tid + t * BLOCK;
      v4f p;
      p.x = vv[t].x * inv;  p.y = vv[t].y * inv;
      p.z = vv[t].z * inv;  p.w = vv[t].w * inv;
      __builtin_nontemporal_store(p, (v4f*)(osm + v4 * 4));
    }

    // Protect ebuf[buf] from the prefetch issued at the top of the next
    // iteration (which targets this same buffer for row + 2*nblocks).
    __syncthreads();
  }
}

extern "C" void kernel_launch(void* const* d_in, const int* in_sizes, int n_in,
                              void* d_out, int out_size, void* d_ws, size_t ws_size,
                              hipStream_t stream) {
  const float* e    = (const float*)d_in[0];
  const float* diag = (const float*)d_in[1];
  const int N = in_sizes[1];            // 4096
  const int B = in_sizes[0] / N;        // 16384
  float* out_softmax = (float*)d_out;
  float* out_scaled  = (float*)d_out + (size_t)B * N;

  int nblocks = (B < GRID_MAX) ? B : GRID_MAX;
  hipLaunchKernelGGL(mnl_masked_softmax_kernel,
                     dim3(nblocks), dim3(BLOCK), 0, stream,
                     e, diag, out_softmax, out_scaled, B, nblocks);
}